// TimeMoEFusion_72335839199356
// MI455X (gfx1250) — compile-verified
//
#include <hip/hip_runtime.h>
#include <math.h>

#define HDIM 768
#define BDIM 32
#define LDIM 256
#define DDIM 2304
#define NROWS (BDIM * LDIM)

typedef float v2f __attribute__((ext_vector_type(2)));
typedef float v8f __attribute__((ext_vector_type(8)));
typedef int   v4i __attribute__((ext_vector_type(4)));

#if defined(__has_builtin)
#  if __has_builtin(__builtin_amdgcn_global_load_async_to_lds_b128)
#    define HAS_ASYNC_LDS 1
#  endif
#  if __has_builtin(__builtin_amdgcn_s_wait_asynccnt)
#    define HAS_WAIT_ASYNC 1
#  endif
#endif

__device__ __forceinline__ void async_copy16(const float* g, float* l)
{
#ifdef HAS_ASYNC_LDS
    __builtin_amdgcn_global_load_async_to_lds_b128(
        (__attribute__((address_space(1))) v4i*)g,
        (__attribute__((address_space(3))) v4i*)l, 0, 0);
#else
    float4 v = *(const float4*)g;
    *(float4*)l = v;
#endif
}

__device__ __forceinline__ void wait_async()
{
#ifdef HAS_ASYNC_LDS
#  ifdef HAS_WAIT_ASYNC
    __builtin_amdgcn_s_wait_asynccnt(0);
#  else
    asm volatile("s_wait_asynccnt 0" ::: "memory");
#  endif
#endif
}

// db4 filters. Analysis = time-reversed decomposition filters (correlation, stride 2).
__constant__ float c_ALO[8] = { 0.23037781330885523f,  0.7148465705525415f,
                                0.6308807679295904f,  -0.02798376941698385f,
                               -0.18703481171888114f,  0.030841381835986965f,
                                0.032883011666982945f,-0.010597401784997278f };
__constant__ float c_AHI[8] = {-0.010597401784997278f,-0.032883011666982945f,
                                0.030841381835986965f, 0.18703481171888114f,
                               -0.02798376941698385f, -0.6308807679295904f,
                                0.7148465705525415f,  -0.23037781330885523f };
// Synthesis = decomposition filters directly (conv_transpose equivalence).
__constant__ float c_SLO[8] = {-0.010597401784997278f, 0.032883011666982945f,
                                0.030841381835986965f,-0.18703481171888114f,
                               -0.02798376941698385f,  0.6308807679295904f,
                                0.7148465705525415f,   0.23037781330885523f };
__constant__ float c_SHI[8] = {-0.23037781330885523f,  0.7148465705525415f,
                               -0.6308807679295904f,  -0.02798376941698385f,
                                0.18703481171888114f,  0.030841381835986965f,
                               -0.032883011666982945f,-0.010597401784997278f };

// ---------------------------------------------------------------------------
// 1) Temporal embedding gather: tid = round(1000*(1-exp(-tdiff/2e7)))
// ---------------------------------------------------------------------------
__global__ __launch_bounds__(256)
void temb_kernel(const int* __restrict__ ts, const float* __restrict__ table,
                 float* __restrict__ temb)
{
    int row = blockIdx.x;            // b*L + l
    int l = row & (LDIM - 1);
    int tcur = ts[row];
    float valid = (tcur != 0) ? 1.0f : 0.0f;
    float td;
    if (l == 0) {
        td = 1.0f;
    } else {
        float d = (float)(tcur - ts[row - 1]);
        td = fmaxf(d, 0.0f);
    }
    td *= valid;
    int tid_ = (int)rintf(1000.0f * (1.0f - expf(-td / 20000000.0f)));
    const float* trow = table + (long)tid_ * HDIM;
    float* orow = temb + (long)row * HDIM;
    for (int h = threadIdx.x; h < HDIM; h += 256) orow[h] = trow[h];
}

// ---------------------------------------------------------------------------
// 2) One DWT level (zero pad, stride 2).  out layout: [s][B*Tout][H]
//    in element (s,b,t,h) at  in + s*sStride + (b*Tin + t)*rowStride + h
// ---------------------------------------------------------------------------
__global__ __launch_bounds__(256)
void dwt_kernel(const float* __restrict__ in, float* __restrict__ lo,
                float* __restrict__ hi, int Tin, int Tout,
                long sStride, int rowStride)
{
    int g = blockIdx.x;                 // (s*B + b)*Tout + t
    int t = g % Tout;
    int sb = g / Tout;
    int b = sb % BDIM;
    int s = sb / BDIM;
    const float* base = in + (long)s * sStride + (long)b * Tin * rowStride;
    long orow = ((long)(s * BDIM + b) * Tout + t) * HDIM;
    for (int h = threadIdx.x; h < HDIM; h += 256) {
        float accL = 0.0f, accH = 0.0f;
#pragma unroll
        for (int k = 0; k < 8; ++k) {
            int ti = 2 * t - 6 + k;     // pad_lo = 6 at every level here
            if (ti >= 0 && ti < Tin) {
                float x = base[(long)ti * rowStride + h];
                accL += x * c_ALO[k];
                accH += x * c_AHI[k];
            }
        }
        lo[orow + h] = accL;
        hi[orow + h] = accH;
    }
}

// ---------------------------------------------------------------------------
// 3) fp32 WMMA GEMM with async global->LDS double buffering.
//    C[M,N] = A[M,K] * W^T, W rows mapped via block structure:
//      Wrow(n) = W + (n / blkRows) * blkStride + (n % blkRows) * K
//    mode 0: out = relu(acc + bias[n])                 -> Cout[M,N]
//    mode 1: fused[n/H][row][n%H] (+)= sigmoid(acc) * A[row][n%H]   (K == H)
//    LDS tiles stored row-major with stride 20 floats (80B): rows 16B-aligned
//    for b128 async writes; stride-20 fragment reads are bank-conflict-free.
// ---------------------------------------------------------------------------
#define LDSW 20

__global__ __launch_bounds__(128)
void wmma_gemm_kernel(const float* __restrict__ A, const float* __restrict__ W,
                      int M, int N, int K, int blkRows, long blkStride,
                      const float* __restrict__ bias, float* __restrict__ Cout,
                      float* __restrict__ fused, int accum, int mode)
{
    __shared__ __align__(16) float As[2][64 * LDSW];
    __shared__ __align__(16) float Bs[2][64 * LDSW];

    const int m0 = blockIdx.y * 64;
    const int n0 = blockIdx.x * 64;
    const int tid = threadIdx.x;
    const int lane = tid & 31;
    const int wv = tid >> 5;       // 4 waves
    const int wm = wv >> 1;        // 0..1 : 32-row half
    const int wn = wv & 1;         // 0..1 : 32-col half
    const int laneHalf = lane >> 4;
    const int lane15 = lane & 15;

    v8f c00 = {}, c01 = {}, c10 = {}, c11 = {};
    const int nK = K >> 4;

    // Stage one 64x16 A tile + 64x16 W tile; each wave moves 2x32 16B chunks each.
    auto issue_tile = [&](int k0, int buf) {
#pragma unroll
        for (int it = 0; it < 2; ++it) {
            int gg = wv * 64 + it * 32 + lane;   // 0..255
            int r = gg >> 2;                     // 0..63
            int c = (gg & 3) * 4;                // 0,4,8,12
            int gr = m0 + r;
            if (gr < M)
                async_copy16(A + (long)gr * K + k0 + c, &As[buf][r * LDSW + c]);
            int n = n0 + r;
            const float* wrow = W + (long)(n / blkRows) * blkStride
                                  + (long)(n % blkRows) * K;
            async_copy16(wrow + k0 + c, &Bs[buf][r * LDSW + c]);
        }
    };

    issue_tile(0, 0);
    for (int i = 0; i < nK; ++i) {
        int cur = i & 1;
        wait_async();          // my async writes into buf `cur` are done
        __syncthreads();       // everyone's writes done; prev compute done
        if (i + 1 < nK) issue_tile((i + 1) * 16, cur ^ 1);

        // Batch all fragments first, then a dense run of 16 WMMAs.
        v2f aF[2][4], bF[2][4];
#pragma unroll
        for (int kk = 0; kk < 4; ++kk) {
            // ISA layout: lanes0-15 hold K=kb,kb+1 ; lanes16-31 hold K=kb+2,kb+3
            int kb = kk * 4 + laneHalf * 2;
            aF[0][kk] = *(const v2f*)&As[cur][(wm * 32 + lane15) * LDSW + kb];
            aF[1][kk] = *(const v2f*)&As[cur][(wm * 32 + 16 + lane15) * LDSW + kb];
            bF[0][kk] = *(const v2f*)&Bs[cur][(wn * 32 + lane15) * LDSW + kb];
            bF[1][kk] = *(const v2f*)&Bs[cur][(wn * 32 + 16 + lane15) * LDSW + kb];
        }
#pragma unroll
        for (int kk = 0; kk < 4; ++kk) {
            c00 = __builtin_amdgcn_wmma_f32_16x16x4_f32(false, aF[0][kk], false, bF[0][kk], (short)0, c00, false, false);
            c01 = __builtin_amdgcn_wmma_f32_16x16x4_f32(false, aF[0][kk], false, bF[1][kk], (short)0, c01, false, false);
            c10 = __builtin_amdgcn_wmma_f32_16x16x4_f32(false, aF[1][kk], false, bF[0][kk], (short)0, c10, false, false);
            c11 = __builtin_amdgcn_wmma_f32_16x16x4_f32(false, aF[1][kk], false, bF[1][kk], (short)0, c11, false, false);
        }
    }

    // Epilogue.  C/D layout: VGPR r -> row r (lanes0-15) / r+8 (lanes16-31), col = lane&15
#pragma unroll
    for (int i = 0; i < 2; ++i) {
#pragma unroll
        for (int j = 0; j < 2; ++j) {
            v8f cc = (i == 0) ? (j == 0 ? c00 : c01) : (j == 0 ? c10 : c11);
#pragma unroll
            for (int r = 0; r < 8; ++r) {
                int row = m0 + wm * 32 + i * 16 + r + laneHalf * 8;
                int col = n0 + wn * 32 + j * 16 + lane15;
                if (row >= M) continue;
                float v = cc[r];
                if (mode == 0) {
                    v += bias[col];
                    v = fmaxf(v, 0.0f);
                    Cout[(long)row * N + col] = v;
                } else {
                    float g = 1.0f / (1.0f + expf(-v));
                    int mi = col / HDIM;
                    int o = col - mi * HDIM;
                    float f = g * A[(long)row * K + o];   // K == HDIM here
                    float* dst = fused + ((long)mi * M + row) * HDIM + o;
                    if (accum) *dst += f; else *dst = f;
                }
            }
        }
    }
}

// ---------------------------------------------------------------------------
// 4) One IDWT level. lo buffer has row-stride loT, only first Tl rows used.
//    Lout = 2*Tl - 6.  y[t] = sum over even j=t-1+k of lo[j/2]*SLO[k]+hi[j/2]*SHI[k]
// ---------------------------------------------------------------------------
__global__ __launch_bounds__(256)
void idwt_kernel(const float* __restrict__ lo, int loT,
                 const float* __restrict__ hi, int Tl,
                 float* __restrict__ out, int Lout)
{
    int g = blockIdx.x;                 // (m*B + b)*Lout + t
    int t = g % Lout;
    int mb = g / Lout;                  // m*B + b
    const float* lbase = lo + (long)mb * loT * HDIM;
    const float* hbase = hi + (long)mb * Tl * HDIM;
    float* obase = out + ((long)mb * Lout + t) * HDIM;
    int jmax = 2 * Tl - 2;
    for (int h = threadIdx.x; h < HDIM; h += 256) {
        float sum = 0.0f;
#pragma unroll
        for (int k = 0; k < 8; ++k) {
            int j = t - 1 + k;
            if (j < 0 || j > jmax || (j & 1)) continue;
            int i = j >> 1;
            sum += lbase[(long)i * HDIM + h] * c_SLO[k]
                 + hbase[(long)i * HDIM + h] * c_SHI[k];
        }
        obase[h] = sum;
    }
}

// ---------------------------------------------------------------------------
// 5) Fusion blend + gate + top-2 MoE combine. One block per token row.
// ---------------------------------------------------------------------------
__global__ __launch_bounds__(256)
void moe_combine_kernel(const float* __restrict__ vec, const float* __restrict__ fused,
                        const float* __restrict__ fw, const float* __restrict__ routing,
                        const float* __restrict__ gate_W, const float* __restrict__ gate_b,
                        const float* __restrict__ shexp, const float* __restrict__ rexp,
                        float* __restrict__ out)
{
    __shared__ float red[4][256];
    __shared__ float sw[2];
    __shared__ int si[2];
    int row = blockIdx.x;
    int tid = threadIdx.x;

    float acc0 = 0.f, acc1 = 0.f, acc2 = 0.f, acc3 = 0.f;
    const float* rrow = routing + (long)row * DDIM;
    for (int d = tid; d < DDIM; d += 256) {
        float r = rrow[d];
        acc0 += r * gate_W[0 * DDIM + d];
        acc1 += r * gate_W[1 * DDIM + d];
        acc2 += r * gate_W[2 * DDIM + d];
        acc3 += r * gate_W[3 * DDIM + d];
    }
    red[0][tid] = acc0; red[1][tid] = acc1; red[2][tid] = acc2; red[3][tid] = acc3;
    __syncthreads();
    for (int sft = 128; sft > 0; sft >>= 1) {
        if (tid < sft) {
            red[0][tid] += red[0][tid + sft];
            red[1][tid] += red[1][tid + sft];
            red[2][tid] += red[2][tid + sft];
            red[3][tid] += red[3][tid + sft];
        }
        __syncthreads();
    }
    if (tid == 0) {
        float lg[4], p[4];
        float mx = -3.4e38f;
        for (int e = 0; e < 4; ++e) { lg[e] = red[e][0] + gate_b[e]; mx = fmaxf(mx, lg[e]); }
        float sum = 0.f;
        for (int e = 0; e < 4; ++e) { p[e] = expf(lg[e] - mx); sum += p[e]; }
        for (int e = 0; e < 4; ++e) p[e] /= sum;
        int i0 = 0; float v0 = p[0];
        for (int e = 1; e < 4; ++e) if (p[e] > v0) { v0 = p[e]; i0 = e; }
        int i1 = -1; float v1 = -3.4e38f;
        for (int e = 0; e < 4; ++e) if (e != i0 && p[e] > v1) { v1 = p[e]; i1 = e; }
        float s2 = fmaxf(v0 + v1, 1e-6f);
        sw[0] = v0 / s2; sw[1] = v1 / s2; si[0] = i0; si[1] = i1;
    }
    __syncthreads();

    float w0 = sw[0], w1 = sw[1];
    int i0 = si[0], i1 = si[1];
    const float fw0 = fw[0], fw1 = fw[1], fw2 = fw[2];
    for (int d = tid; d < DDIM; d += 256) {
        int s = d / HDIM;
        int h = d - s * HDIM;
        float fws = (s == 0) ? fw0 : (s == 1 ? fw1 : fw2);
        float x = vec[(long)row * DDIM + d];
        float fu = fused[((long)s * NROWS + row) * HDIM + h];
        float enh = x + (fu - x) * fws;
        float coef = shexp[d] + w0 * rexp[(long)i0 * DDIM + d]
                              + w1 * rexp[(long)i1 * DDIM + d];
        out[(long)s * ((long)NROWS * HDIM) + (long)row * HDIM + h] = enh * coef;
    }
}

// ---------------------------------------------------------------------------
extern "C" void kernel_launch(void* const* d_in, const int* in_sizes, int n_in,
                              void* d_out, int out_size, void* d_ws, size_t ws_size,
                              hipStream_t stream)
{
    const float* vector = (const float*)d_in[0];
    const int*   tstamp = (const int*)d_in[1];
    const float* low_W  = (const float*)d_in[2];
    const float* high_W = (const float*)d_in[3];
    const float* fw     = (const float*)d_in[4];
    const float* table  = (const float*)d_in[5];
    const float* proj_W = (const float*)d_in[6];
    const float* proj_b = (const float*)d_in[7];
    const float* gate_W = (const float*)d_in[8];
    const float* gate_b = (const float*)d_in[9];
    const float* shexp  = (const float*)d_in[10];
    const float* rexp   = (const float*)d_in[11];
    float* out = (float*)d_out;

    const int T[5]  = {256, 131, 69, 38, 22};   // T[lv] = coeff length at level lv
    const int RT[4] = {38, 70, 132, 256};       // reconstruction lengths

    float* ws = (float*)d_ws;
    long off = 0;
    auto alloc = [&](long n) { float* p = ws + off; off += n; return p; };

    float* temb    = alloc((long)NROWS * HDIM);
    float* routing = alloc((long)NROWS * DDIM);
    float* lo[5];  float* hi[5];
    lo[0] = nullptr; hi[0] = nullptr;
    for (int lv = 1; lv <= 4; ++lv) {
        lo[lv] = alloc(3L * BDIM * T[lv] * HDIM);
        hi[lv] = alloc(3L * BDIM * T[lv] * HDIM);
    }
    float* flo = alloc(3L * BDIM * T[4] * HDIM);
    float* fh[5]; fh[0] = nullptr;
    for (int lv = 1; lv <= 4; ++lv) fh[lv] = alloc(3L * BDIM * T[lv] * HDIM);
    float* rec[4];
    for (int i = 0; i < 4; ++i) rec[i] = alloc(3L * BDIM * RT[i] * HDIM);
    if ((size_t)off * sizeof(float) > ws_size) return;  // scratch too small

    // 1) temporal embedding
    temb_kernel<<<NROWS, 256, 0, stream>>>(tstamp, table, temb);

    // 2) DWT: level 1 reads straight from vector ([B,L,3H] strided)
    dwt_kernel<<<3 * BDIM * T[1], 256, 0, stream>>>(vector, lo[1], hi[1],
                                                    T[0], T[1], (long)HDIM, DDIM);
    for (int lv = 2; lv <= 4; ++lv)
        dwt_kernel<<<3 * BDIM * T[lv], 256, 0, stream>>>(lo[lv - 1], lo[lv], hi[lv],
            T[lv - 1], T[lv], (long)BDIM * T[lv - 1] * HDIM, HDIM);

    // 3a) routing = relu(temb @ proj_W^T + proj_b)
    dim3 blk(128);
    wmma_gemm_kernel<<<dim3(DDIM / 64, (NROWS + 63) / 64), blk, 0, stream>>>(
        temb, proj_W, NROWS, DDIM, HDIM, DDIM, 0L, proj_b, routing,
        nullptr, 0, 0);

    // 3b) gating GEMMs: per band, per source modality s, all 3 outputs m fused
    const long HH2 = (long)HDIM * HDIM;
    for (int s = 0; s < 3; ++s) {
        int M = BDIM * T[4];
        wmma_gemm_kernel<<<dim3(DDIM / 64, (M + 63) / 64), blk, 0, stream>>>(
            lo[4] + (long)s * BDIM * T[4] * HDIM, low_W + (long)s * HH2,
            M, DDIM, HDIM, HDIM, 3 * HH2, nullptr, nullptr, flo, (s > 0), 1);
        for (int k = 0; k < 4; ++k) {
            int Mh = BDIM * T[k + 1];
            wmma_gemm_kernel<<<dim3(DDIM / 64, (Mh + 63) / 64), blk, 0, stream>>>(
                hi[k + 1] + (long)s * BDIM * T[k + 1] * HDIM,
                high_W + ((long)k * 3 + s) * HH2,
                Mh, DDIM, HDIM, HDIM, 12 * HH2, nullptr, nullptr,
                fh[k + 1], (s > 0), 1);
        }
    }

    // 4) IDWT: lo cropped to hi length each step (crop via Tl < loT)
    idwt_kernel<<<3 * BDIM * RT[0], 256, 0, stream>>>(flo,    T[4],  fh[4], T[4], rec[0], RT[0]);
    idwt_kernel<<<3 * BDIM * RT[1], 256, 0, stream>>>(rec[0], RT[0], fh[3], T[3], rec[1], RT[1]);
    idwt_kernel<<<3 * BDIM * RT[2], 256, 0, stream>>>(rec[1], RT[1], fh[2], T[2], rec[2], RT[2]);
    idwt_kernel<<<3 * BDIM * RT[3], 256, 0, stream>>>(rec[2], RT[2], fh[1], T[1], rec[3], RT[3]);

    // 5) blend + gate + top-2 MoE
    moe_combine_kernel<<<NROWS, 256, 0, stream>>>(vector, rec[3], fw, routing,
                                                  gate_W, gate_b, shexp, rexp, out);
}